// Texture_46222438040249
// MI455X (gfx1250) — compile-verified
//
#include <hip/hip_runtime.h>
#include <stdint.h>

#define HOWO (512 * 512)

__device__ __forceinline__ void fma4(float4& a, const float4 v, float w) {
    a.x = fmaf(w, v.x, a.x);
    a.y = fmaf(w, v.y, a.y);
    a.z = fmaf(w, v.z, a.z);
    a.w = fmaf(w, v.w, a.w);
}

struct BilinW {
    int cx0, cx1, cy0, cy1;
    float w00, w01, w10, w11;
};

__device__ __forceinline__ BilinW bw_calc(int S, float fx, float fy) {
    // grid = x*2-1 ; ix = (grid+1)*0.5*(S-1) == x*(S-1)
    float ix = fx * (float)(S - 1);
    float iy = fy * (float)(S - 1);
    float ix0f = floorf(ix), iy0f = floorf(iy);
    float wx1 = ix - ix0f, wy1 = iy - iy0f;
    float wx0 = 1.0f - wx1, wy0 = 1.0f - wy1;
    int ix0 = (int)ix0f, iy0 = (int)iy0f;
    int ix1 = ix0 + 1, iy1 = iy0 + 1;
    bool vx0 = (unsigned)ix0 < (unsigned)S;
    bool vx1 = (unsigned)ix1 < (unsigned)S;
    bool vy0 = (unsigned)iy0 < (unsigned)S;
    bool vy1 = (unsigned)iy1 < (unsigned)S;
    BilinW b;
    b.cx0 = min(max(ix0, 0), S - 1);
    b.cx1 = min(max(ix1, 0), S - 1);
    b.cy0 = min(max(iy0, 0), S - 1);
    b.cy1 = min(max(iy1, 0), S - 1);
    b.w00 = (vy0 && vx0) ? wy0 * wx0 : 0.0f;
    b.w01 = (vy0 && vx1) ? wy0 * wx1 : 0.0f;
    b.w10 = (vy1 && vx0) ? wy1 * wx0 : 0.0f;
    b.w11 = (vy1 && vx1) ? wy1 * wx1 : 0.0f;
    return b;
}

// channel-last layer sample: issue all 16 global_load_b128 (4 corners x 64B)
// up-front for max outstanding L2 gathers, then 64 FMAs.
__device__ __forceinline__ void sample_cl(const float* __restrict__ t, int S,
                                          float fx, float fy, float4* a) {
    BilinW b = bw_calc(S, fx, fy);
    const float4* p00 = (const float4*)(t + (((size_t)b.cy0 * (size_t)S + (size_t)b.cx0) << 4));
    const float4* p01 = (const float4*)(t + (((size_t)b.cy0 * (size_t)S + (size_t)b.cx1) << 4));
    const float4* p10 = (const float4*)(t + (((size_t)b.cy1 * (size_t)S + (size_t)b.cx0) << 4));
    const float4* p11 = (const float4*)(t + (((size_t)b.cy1 * (size_t)S + (size_t)b.cx1) << 4));
    float4 q00a = p00[0], q00b = p00[1], q00c = p00[2], q00d = p00[3];
    float4 q01a = p01[0], q01b = p01[1], q01c = p01[2], q01d = p01[3];
    float4 q10a = p10[0], q10b = p10[1], q10c = p10[2], q10d = p10[3];
    float4 q11a = p11[0], q11b = p11[1], q11c = p11[2], q11d = p11[3];
    fma4(a[0], q00a, b.w00); fma4(a[1], q00b, b.w00);
    fma4(a[2], q00c, b.w00); fma4(a[3], q00d, b.w00);
    fma4(a[0], q01a, b.w01); fma4(a[1], q01b, b.w01);
    fma4(a[2], q01c, b.w01); fma4(a[3], q01d, b.w01);
    fma4(a[0], q10a, b.w10); fma4(a[1], q10b, b.w10);
    fma4(a[2], q10c, b.w10); fma4(a[3], q10d, b.w10);
    fma4(a[0], q11a, b.w11); fma4(a[1], q11b, b.w11);
    fma4(a[2], q11c, b.w11); fma4(a[3], q11d, b.w11);
}

// fallback: channel-major [C][S][S] gather (used only if workspace too small)
__device__ __forceinline__ void sample_cf(const float* __restrict__ t, int S,
                                          float fx, float fy, float* acc) {
    BilinW b = bw_calc(S, fx, fy);
    size_t HW = (size_t)S * (size_t)S;
    size_t o00 = (size_t)b.cy0 * S + b.cx0;
    size_t o01 = (size_t)b.cy0 * S + b.cx1;
    size_t o10 = (size_t)b.cy1 * S + b.cx0;
    size_t o11 = (size_t)b.cy1 * S + b.cx1;
#pragma unroll
    for (int c = 0; c < 16; ++c) {
        const float* tc = t + (size_t)c * HW;
        acc[c] += b.w00 * tc[o00] + b.w01 * tc[o01] + b.w10 * tc[o10] + b.w11 * tc[o11];
    }
}

// prefetch the 2x2 footprint rows for a lower mip level (global_prefetch_b8)
__device__ __forceinline__ void prefetch_cl(const float* __restrict__ t, int S,
                                            float fx, float fy) {
    int ix0 = (int)floorf(fx * (float)(S - 1));
    int iy0 = (int)floorf(fy * (float)(S - 1));
    ix0 = min(max(ix0, 0), S - 1);
    iy0 = min(max(iy0, 0), S - 1);
    int iy1 = min(iy0 + 1, S - 1);
    __builtin_prefetch((const char*)(t + (((size_t)iy0 * S + ix0) << 4)), 0, 3);
    __builtin_prefetch((const char*)(t + (((size_t)iy1 * S + ix0) << 4)), 0, 3);
}

// one-time layout change: [16][S][S] -> [S][S][16].
// Source is streamed exactly once -> non-temporal loads (don't evict the
// gather-hot data from L2). Destination is the gather target -> regular
// stores so the transposed texels become/stay L2-resident (192MB L2 holds
// all 85MB of transposed texture).
__global__ __launch_bounds__(256) void transpose_cl_kernel(const float* __restrict__ src,
                                                           float* __restrict__ dst, int S) {
    const size_t HW = (size_t)S * (size_t)S;
    size_t idx = (size_t)blockIdx.x * 256 + threadIdx.x;
    if (idx >= HW) return;
    float v[16];
#pragma unroll
    for (int c = 0; c < 16; ++c) v[c] = __builtin_nontemporal_load(src + (size_t)c * HW + idx);
    float4* d = (float4*)(dst + (idx << 4));
    d[0] = make_float4(v[0], v[1], v[2], v[3]);
    d[1] = make_float4(v[4], v[5], v[6], v[7]);
    d[2] = make_float4(v[8], v[9], v[10], v[11]);
    d[3] = make_float4(v[12], v[13], v[14], v[15]);
}

template <bool CL>
__global__ __launch_bounds__(256) void tex_sample_kernel(
    const float* __restrict__ x, const float* __restrict__ msk,
    const float* __restrict__ t1, const float* __restrict__ t2,
    const float* __restrict__ t3, const float* __restrict__ t4,
    float* __restrict__ out) {
    __shared__ float sg[256 * 2];
    __shared__ float sm[256];
    const int tid = threadIdx.x;
    const int idx = blockIdx.x * 256 + tid;

    // CDNA5 async global->LDS staging of this lane's grid coords + mask.
    // Per-lane self-copy: wave-local s_wait_asynccnt suffices (no barrier).
    {
        unsigned lds_g = (unsigned)(uintptr_t)&sg[tid * 2];
        unsigned lds_m = (unsigned)(uintptr_t)&sm[tid];
        const float* gp = x + (size_t)idx * 2;
        const float* mp = msk + idx;
        asm volatile("global_load_async_to_lds_b64 %0, %1, off"
                     :: "v"(lds_g), "v"(gp) : "memory");
        asm volatile("global_load_async_to_lds_b32 %0, %1, off"
                     :: "v"(lds_m), "v"(mp) : "memory");
        asm volatile("s_wait_asynccnt 0x0" ::: "memory");
    }
    float fx = sg[tid * 2 + 0];
    float fy = sg[tid * 2 + 1];
    float m = sm[tid];

    float4 acc[4] = {};
    if constexpr (CL) {
        // hide L2 gather latency of lower mips behind level-1 FMAs
        prefetch_cl(t2, 512, fx, fy);
        prefetch_cl(t3, 256, fx, fy);
        prefetch_cl(t4, 128, fx, fy);
        sample_cl(t1, 1024, fx, fy, acc);
        sample_cl(t2, 512, fx, fy, acc);
        sample_cl(t3, 256, fx, fy, acc);
        sample_cl(t4, 128, fx, fy, acc);
    } else {
        float* a = (float*)acc;
        sample_cf(t1, 1024, fx, fy, a);
        sample_cf(t2, 512, fx, fy, a);
        sample_cf(t3, 256, fx, fy, a);
        sample_cf(t4, 128, fx, fy, a);
    }

    const int b = idx >> 18;          // HoWo = 2^18
    const int p = idx & (HOWO - 1);
    float* ob = out + (((size_t)b * 16) << 18) + p;
    const float* a = (const float*)acc;
    // Output is written once and never re-read: non-temporal stores keep the
    // 67MB result from evicting the gather-hot textures out of L2.
#pragma unroll
    for (int c = 0; c < 16; ++c)
        __builtin_nontemporal_store(a[c] * m, ob + ((size_t)c << 18));
}

extern "C" void kernel_launch(void* const* d_in, const int* in_sizes, int n_in,
                              void* d_out, int out_size, void* d_ws, size_t ws_size,
                              hipStream_t stream) {
    const float* x   = (const float*)d_in[0];  // [4,512,512,2]
    const float* msk = (const float*)d_in[1];  // [4,512,512]
    const float* l1  = (const float*)d_in[2];  // [16,1024,1024]
    const float* l2  = (const float*)d_in[3];  // [16,512,512]
    const float* l3  = (const float*)d_in[4];  // [16,256,256]
    const float* l4  = (const float*)d_in[5];  // [16,128,128]
    float* out = (float*)d_out;

    const size_t n1 = 1024 * 1024, n2 = 512 * 512, n3 = 256 * 256, n4 = 128 * 128;
    const size_t need = (n1 + n2 + n3 + n4) * 16 * sizeof(float);  // ~85 MB

    const int npts = 4 * HOWO;          // 1,048,576 points
    const int blocks = npts / 256;      // 4096

    if (ws_size >= need) {
        float* t1 = (float*)d_ws;
        float* t2 = t1 + n1 * 16;
        float* t3 = t2 + n2 * 16;
        float* t4 = t3 + n3 * 16;
        transpose_cl_kernel<<<(int)(n1 / 256), 256, 0, stream>>>(l1, t1, 1024);
        transpose_cl_kernel<<<(int)(n2 / 256), 256, 0, stream>>>(l2, t2, 512);
        transpose_cl_kernel<<<(int)(n3 / 256), 256, 0, stream>>>(l3, t3, 256);
        transpose_cl_kernel<<<(int)(n4 / 256), 256, 0, stream>>>(l4, t4, 128);
        tex_sample_kernel<true><<<blocks, 256, 0, stream>>>(x, msk, t1, t2, t3, t4, out);
    } else {
        tex_sample_kernel<false><<<blocks, 256, 0, stream>>>(x, msk, l1, l2, l3, l4, out);
    }
}